// BertForMultilabelNER_30812095382008
// MI455X (gfx1250) — compile-verified
//
#include <hip/hip_runtime.h>
#include <hip/hip_bf16.h>

typedef __attribute__((ext_vector_type(2))) float v2f;
typedef __attribute__((ext_vector_type(8))) float v8f;

#define BB 64
#define SS 512
#define WW 256
#define DD 768
#define NN 96          // A*3 output columns
#define NT 6           // N tiles of 16
#define OUT_PER_A (BB * WW * 3)   // 49152

// One wave computes a 16(M) x 96(N) slab of the GEMM:
//   C[m, n] = sum_k pooled[m, k] * clsw[n, k] + clsb[n]
// where pooled[m, :] = seq[b, word_idxs[b, w], :], m = b*W + w.
// Exact-fp32 V_WMMA_F32_16X16X4_F32, K-loop of 192 steps, A reused 6x per step.
__global__ __launch_bounds__(64) void ner_head_wmma_f32(
    const float* __restrict__ seq,   // [B, S, D] fp32
    const int* __restrict__ widx,    // [B, W] int32
    const float* __restrict__ clsw,  // [A, 3, D] == row-major [96][768]
    const float* __restrict__ clsb,  // [A, 3] == [96]
    float* __restrict__ out)         // [A, B, W, 3]
{
    const int lane = threadIdx.x & 31;
    const int wave = threadIdx.x >> 5;
    const int mtile = blockIdx.x * 2 + wave;   // 0..1023
    const int mlo = lane & 15;                 // row (for A) / col (for B) within tile
    const int koff = (lane >> 4) << 1;         // 0 or 2: which K-pair this half-wave owns

    // --- gather: resolve this lane's A row pointer ---
    const int m = mtile * 16 + mlo;            // global pooled-row id
    const int b = m >> 8;                      // m / W   (W = 256)
    const int w = m & 255;                     // m % W
    const int srow = widx[b * WW + w];         // subword index into S
    const float* __restrict__ arow = seq + ((size_t)b * SS + (size_t)srow) * DD + koff;

    // --- B: lane reads column n = t*16 + mlo of clsw (row-major [96][768]) ---
    const float* __restrict__ brow = clsw + (size_t)mlo * DD + koff;

    v8f acc[NT] = {};   // 6 x (16x16 f32) accumulators

    for (int k0 = 0; k0 < DD; k0 += 16) {
        // prefetch next 128B line of the gathered row, WGP scope (locality=3):
        // pulls into all cache levels; speculative so tail overrun is safe
        __builtin_prefetch(arow + k0 + 32, 0, 3);
#pragma unroll
        for (int k = k0; k < k0 + 16; k += 4) {
            const v2f a = *(const v2f*)(arow + k);
#pragma unroll
            for (int t = 0; t < NT; ++t) {
                const v2f bv = *(const v2f*)(brow + (size_t)(t * 16) * DD + k);
                // 8 args: (neg_a, A, neg_b, B, c_mod, C, reuse_a, reuse_b)
                acc[t] = __builtin_amdgcn_wmma_f32_16x16x4_f32(
                    false, a, false, bv, (short)0, acc[t], false, false);
            }
        }
    }

    // --- epilogue: bias + scatter into [A, B, W, 3] ---
    // C/D layout: VGPR r -> M = r + 8*(lane>=16), N = lane&15
    const int mhalf = (lane >> 4) << 3;        // 0 or 8
#pragma unroll
    for (int t = 0; t < NT; ++t) {
        const int ng = t * 16 + mlo;           // global output column, 0..95
        const int ai = ng / 3;                 // attribute a
        const int c = ng - ai * 3;             // class c
        const float bias = clsb[ng];
#pragma unroll
        for (int r = 0; r < 8; ++r) {
            const int mrow = mtile * 16 + mhalf + r;
            out[(size_t)ai * OUT_PER_A + (size_t)mrow * 3 + c] = acc[t][r] + bias;
        }
    }
}

extern "C" void kernel_launch(void* const* d_in, const int* in_sizes, int n_in,
                              void* d_out, int out_size, void* d_ws, size_t ws_size,
                              hipStream_t stream) {
    (void)in_sizes; (void)n_in; (void)out_size; (void)d_ws; (void)ws_size;
    const float* seq  = (const float*)d_in[0];  // [B,S,D] fp32
    const int*   widx = (const int*)d_in[1];    // [B,W] int32
    const float* clsw = (const float*)d_in[2];  // [A,3,D] fp32
    const float* clsb = (const float*)d_in[3];  // [A,3] fp32
    float* out = (float*)d_out;                 // [A,B,W,3] fp32

    // 1024 M-tiles, one per wave; 2 waves per 64-thread block -> 512 blocks
    // (no LDS / no barriers, so small blocks maximize spread across WGPs)
    ner_head_wmma_f32<<<512, 64, 0, stream>>>(seq, widx, clsw, clsb, out);
}